// DLEM_84980222919452
// MI455X (gfx1250) — compile-verified
//
#include <hip/hip_runtime.h>
#include <hip/hip_bf16.h>

typedef float v2f __attribute__((ext_vector_type(2)));
typedef float v8f __attribute__((ext_vector_type(8)));
typedef int   v4i __attribute__((vector_size(16)));

#define N_DIM 2048
#define B_DIM 8
#define CHN   10
#define NDIAG 64
#define TS    64
#define KSPLIT 4
#define NT    ((N_DIM / TS) / KSPLIT)      // 8 column tiles per block
#define TOT_PER_B (NDIAG*(N_DIM-1) - (NDIAG*(NDIAG+1))/2)   // 128928

#ifndef __has_builtin
#define __has_builtin(x) 0
#endif

#if __has_builtin(__builtin_amdgcn_global_load_async_to_lds_b128)
#define HAVE_ASYNC_LDS 1
#else
#define HAVE_ASYNC_LDS 0
#endif

#if HAVE_ASYNC_LDS
__device__ __forceinline__ void async_copy16(float* lds, const float* g) {
  // 16B global -> LDS, tracked by ASYNCcnt (per-lane addresses).
  // Builtin expects (AS1 v4i*, AS3 v4i*, int offset, int cpol).
  v4i* gg = (v4i*)(float*)g;                 // drop const, reinterpret (generic)
  v4i* ll = (v4i*)lds;                       // reinterpret (generic)
  __builtin_amdgcn_global_load_async_to_lds_b128(
      (__attribute__((address_space(1))) v4i*)gg,
      (__attribute__((address_space(3))) v4i*)ll, 0, 0);
}
#endif

#if __has_builtin(__builtin_amdgcn_s_wait_asynccnt)
#define WAIT_ASYNC(n) __builtin_amdgcn_s_wait_asynccnt(n)
#else
#define WAIT_ASYNC(n) asm volatile("s_wait_asynccnt %0" :: "i"(n))
#endif

// Per-wave async instructions per tile: 8 (Xs) + 2 (Wrow)
#define GASYNC 10

// ---------------------------------------------------------------------------
// Kernel 1: fused row-dot / col-dot GEMM, single pass over x (134 MB).
// Block = 4 waves, owns a 64-row strip, loops over NT=8 column tiles with a
// double-buffered LDS tile filled by GLOBAL_LOAD_ASYNC_TO_LDS_B128.
// RD accumulates in registers across the K loop (4 atomic contribs via
// KSPLIT); CD partials per tile via f32 atomics (L2-resident 1MB buffer).
// ---------------------------------------------------------------------------
__global__ __launch_bounds__(128) void k_gemm_lr(
    const float* __restrict__ x, const float* __restrict__ wl,
    const float* __restrict__ wr, float* __restrict__ RD, float* __restrict__ CD)
{
  __shared__ __align__(16) float Xs[2][TS][TS + 4];     // stride 68: 4-bank skew, rows 16B-aligned
  __shared__ __align__(16) float Wrow[2][16][TS + 4];   // [ch][K], ch15 = dup of ch14 (discarded col)
  __shared__ float Wcol[TS][16];                        // [K=row][ch], fixed per block

  const int b        = blockIdx.z;
  const int rowBase  = blockIdx.x * TS;
  const int colStart = blockIdx.y * (TS * NT);
  const int t    = threadIdx.x;
  const int lane = t & 31;
  const int wave = t >> 5;
  const float* xb = x + (size_t)b * N_DIM * N_DIM;

  // Wcol: once per block (depends only on rowBase)
  for (int i = t; i < TS * 16; i += 128) {
    int K = i >> 4, ch = i & 15;
    float c = 0.f;
    if (ch < 15) {
      int cc = ch / 3, kw = ch - cc * 3;
      c = wr[((size_t)cc * N_DIM + rowBase + K) * 3 + kw];
    }
    Wcol[K][ch] = c;
  }

  // ---- tile fetch (async double-buffer; sync fallback) --------------------
  auto issue_tile = [&](int buf, int colBase) {
#if HAVE_ASYNC_LDS
    #pragma unroll
    for (int i = 0; i < 8; ++i) {
      int idx = t + i * 128;
      int r   = idx >> 4;
      int c4  = (idx & 15) * 4;
      async_copy16(&Xs[buf][r][c4],
                   xb + (size_t)(rowBase + r) * N_DIM + colBase + c4);
    }
    #pragma unroll
    for (int i = 0; i < 2; ++i) {
      int idx = t + i * 128;          // 0..255 over 16 ch x 16 float4
      int ch  = idx >> 4;
      int c4  = (idx & 15) * 4;
      int chc = (ch < 15) ? ch : 14;  // ch15 feeds only the discarded D column
      async_copy16(&Wrow[buf][ch][c4],
                   wl + (size_t)chc * N_DIM + colBase + c4);
    }
#else
    #pragma unroll
    for (int i = 0; i < 8; ++i) {
      int idx = t + i * 128;
      int r   = idx >> 4;
      int c4  = (idx & 15) * 4;
      const float4 v = *(const float4*)(xb + (size_t)(rowBase + r) * N_DIM + colBase + c4);
      Xs[buf][r][c4+0]=v.x; Xs[buf][r][c4+1]=v.y; Xs[buf][r][c4+2]=v.z; Xs[buf][r][c4+3]=v.w;
    }
    #pragma unroll
    for (int i = 0; i < 2; ++i) {
      int idx = t + i * 128;
      int ch  = idx >> 4;
      int c4  = (idx & 15) * 4;
      int chc = (ch < 15) ? ch : 14;
      const float4 v = *(const float4*)(wl + (size_t)chc * N_DIM + colBase + c4);
      Wrow[buf][ch][c4+0]=v.x; Wrow[buf][ch][c4+1]=v.y; Wrow[buf][ch][c4+2]=v.z; Wrow[buf][ch][c4+3]=v.w;
    }
#endif
  };

  issue_tile(0, colStart);

  // f32 WMMA operand mapping (ISA 16x16x4): A VGPR0 = K{0|2 by lane half},
  // VGPR1 = K{1|3}; C/D VGPR i = row (i + 8*half), col = lane&15.
  const int half2 = (lane >> 4) * 2;
  const int l16   = lane & 15;
  const int strip = wave * 16 + l16;

  v8f accR = {};
  for (int ct = 0; ct < NT; ++ct) {
    const int buf     = ct & 1;
    const int colBase = colStart + ct * TS;

    if (ct + 1 < NT) {                 // overlap next tile's HBM fetch
      issue_tile(buf ^ 1, colStart + (ct + 1) * TS);
      WAIT_ASYNC(GASYNC);              // older batch (current buf) retired
    } else {
      WAIT_ASYNC(0);
    }
    __syncthreads();

    #pragma unroll 4
    for (int kk = 0; kk < TS; kk += 4) {      // RD: A[M=row][K=col]
      int k0 = kk + half2;
      v2f a, w;
      a.x = Xs[buf][strip][k0];   a.y = Xs[buf][strip][k0 + 1];
      w.x = Wrow[buf][l16][k0];   w.y = Wrow[buf][l16][k0 + 1];
      accR = __builtin_amdgcn_wmma_f32_16x16x4_f32(false, a, false, w, (short)0,
                                                   accR, false, false);
    }

    v8f accC = {};
    #pragma unroll 4
    for (int kk = 0; kk < TS; kk += 4) {      // CD: A[M=col][K=row] (transposed read)
      int k0 = kk + half2;
      v2f a, w;
      a.x = Xs[buf][k0][strip];   a.y = Xs[buf][k0 + 1][strip];
      w.x = Wcol[k0][l16];        w.y = Wcol[k0 + 1][l16];
      accC = __builtin_amdgcn_wmma_f32_16x16x4_f32(false, a, false, w, (short)0,
                                                   accC, false, false);
    }

    if (l16 < 15) {
      #pragma unroll
      for (int i = 0; i < 8; ++i) {
        int M = i + (half2 << 2);
        atomicAdd(&CD[((size_t)b * N_DIM + colBase + wave * 16 + M) * 16 + l16], accC[i]);
      }
    }
    __syncthreads();                   // readers done before buf is refilled
  }

  if (l16 < 15) {
    #pragma unroll
    for (int i = 0; i < 8; ++i) {
      int M = i + (half2 << 2);
      atomicAdd(&RD[((size_t)b * N_DIM + rowBase + wave * 16 + M) * 16 + l16], accR[i]);
    }
  }
}

// ---------------------------------------------------------------------------
// Kernel 2: assemble lr from RD/CD (+kh shifts, biases), 6 tiny conv layers
// in an LDS ping-pong (512-col chunks, halo 8 > 6 layers of shrink), then the
// 1x1 sigmoid head -> mleft/mright.
// ---------------------------------------------------------------------------
#define CHUNK 512
#define HALO  8
#define CW    (CHUNK + 2 * HALO)

__global__ __launch_bounds__(256) void k_mlp(
    const float* __restrict__ RD, const float* __restrict__ CD,
    const float* __restrict__ bl, const float* __restrict__ br,
    const float* __restrict__ wA, const float* __restrict__ bA,
    const float* __restrict__ wB, const float* __restrict__ bB,
    const float* __restrict__ wT, const float* __restrict__ bT,
    const float* __restrict__ wm, const float* __restrict__ bm,
    float* __restrict__ mleft, float* __restrict__ mright)
{
  __shared__ float h0[CHN][CW];
  __shared__ float h1[CHN][CW];
  __shared__ float W[3][CHN * CHN * 3];
  __shared__ float Bb[3][CHN];
  __shared__ float wms[2 * CHN];
  __shared__ float bms[2];

  const int b     = blockIdx.y;
  const int cbase = blockIdx.x * CHUNK;
  const int t     = threadIdx.x;

  for (int i = t; i < CHN * CHN * 3; i += 256) {
    W[0][i] = wA[i];
    W[1][i] = wB[i];
    int o = i / (CHN * 3);
    int r = i - o * (CHN * 3);
    int c = r / 3, k = r - c * 3;
    W[2][i] = wT[(c * CHN + o) * 3 + (2 - k)];   // wTc = flip(swapaxes(wT,0,1),-1)
  }
  if (t < CHN)     { Bb[0][t] = bA[t]; Bb[1][t] = bB[t]; Bb[2][t] = bT[t]; }
  if (t < 2 * CHN) wms[t] = wm[t];
  if (t < 2)       bms[t] = bm[t];

  for (int i = t; i < CHN * CW; i += 256) {
    int ch  = i / CW;
    int w   = i - ch * CW;
    int col = cbase - HALO + w;
    float v = 0.f;
    if (col >= 0 && col < N_DIM) {
      if (ch < 5) {
        float s = bl[ch];
        const float* rd = RD + (size_t)b * N_DIM * 16;
        if (col > 0)         s += rd[(size_t)(col - 1) * 16 + ch * 3 + 0];
        s += rd[(size_t)col * 16 + ch * 3 + 1];
        if (col + 1 < N_DIM) s += rd[(size_t)(col + 1) * 16 + ch * 3 + 2];
        v = s;
      } else {
        int c = ch - 5;
        float s = br[c];
        const float* cd = CD + (size_t)b * N_DIM * 16;
        if (col > 0)         s += cd[(size_t)(col - 1) * 16 + c * 3 + 0];
        s += cd[(size_t)col * 16 + c * 3 + 1];
        if (col + 1 < N_DIM) s += cd[(size_t)(col + 1) * 16 + c * 3 + 2];
        v = s;
      }
    }
    h0[ch][w] = v;
  }
  __syncthreads();

  float* cur = &h0[0][0];
  float* nxt = &h1[0][0];
  for (int layer = 0; layer < 6; ++layer) {
    const int wi = (layer == 0) ? 0 : ((layer < 3) ? 1 : 2);
    for (int i = t; i < CHN * CW; i += 256) {
      int o   = i / CW;
      int w   = i - o * CW;
      int col = cbase - HALO + w;
      float v = 0.f;
      if (w > 0 && w < CW - 1 && col >= 0 && col < N_DIM) {
        float s = Bb[wi][o];
        #pragma unroll
        for (int c = 0; c < CHN; ++c) {
          const float* wk = &W[wi][(o * CHN + c) * 3];
          const float* hc = cur + c * CW + w;
          s = fmaf(wk[0], hc[-1], s);
          s = fmaf(wk[1], hc[0],  s);
          s = fmaf(wk[2], hc[1],  s);
        }
        v = fmaxf(s, 0.f);
      }
      nxt[o * CW + w] = v;
    }
    __syncthreads();
    float* tmp = cur; cur = nxt; nxt = tmp;
  }

  for (int i = t; i < CHUNK; i += 256) {
    int w   = HALO + i;
    int col = cbase + i;
    float s0 = bms[0], s1 = bms[1];
    #pragma unroll
    for (int c = 0; c < CHN; ++c) {
      float hv = cur[c * CW + w];
      s0 = fmaf(wms[c],       hv, s0);
      s1 = fmaf(wms[CHN + c], hv, s1);
    }
    mleft [(size_t)b * N_DIM + col] = 1.f / (1.f + __expf(-s0));
    mright[(size_t)b * N_DIM + col] = 1.f / (1.f + __expf(-s1));
  }
}

// ---------------------------------------------------------------------------
// Kernel 3a: per-diagonal log mass ratios + per-d sum (for the mean).
// ---------------------------------------------------------------------------
__global__ __launch_bounds__(256) void k_diag(
    const float* __restrict__ x, const float* __restrict__ mleft,
    const float* __restrict__ mright, float* __restrict__ out,
    float* __restrict__ sums)
{
  const int d = blockIdx.x + 1;           // start=1 .. stop-1=64
  const int b = blockIdx.y;
  const int L = N_DIM - d - 1;
  const size_t off = (size_t)(d - 1) * (N_DIM - 1) - (size_t)(d - 1) * d / 2;

  const float* xb = x + (size_t)b * N_DIM * N_DIM;
  const float* lf = mleft  + (size_t)b * N_DIM;
  const float* rt = mright + (size_t)b * N_DIM;
  float* ob = out + (size_t)b * TOT_PER_B + off;

  float lsum = 0.f;
  for (int k = threadIdx.x; k < L; k += 256) {
    if (k + 512 < L)   // strided diagonal walk: prefetch ahead
      __builtin_prefetch(xb + (size_t)(k + 512) * N_DIM + (k + 512) + d, 0, 0);
    float c0 = __expf(xb[(size_t)k * N_DIM + k + d]);
    float c1 = __expf(xb[(size_t)(k + 1) * N_DIM + k + 1 + d]);
    float mi = c1 * rt[k + 1] + c0 * lf[k + d];
    float mo = rt[k] + lf[k + d + 1];
    float p  = __logf(mi / mo);
    ob[k] = p;
    lsum += p;
  }

  __shared__ float red[256];
  red[threadIdx.x] = lsum;
  __syncthreads();
  for (int s = 128; s > 0; s >>= 1) {
    if (threadIdx.x < s) red[threadIdx.x] += red[threadIdx.x + s];
    __syncthreads();
  }
  if (threadIdx.x == 0) atomicAdd(&sums[d - 1], red[0]);
}

// Kernel 3b: subtract per-d mean (over all B*L elements).
__global__ __launch_bounds__(256) void k_center(
    float* __restrict__ out, const float* __restrict__ sums)
{
  const int d = blockIdx.x + 1;
  const int b = blockIdx.y;
  const int L = N_DIM - d - 1;
  const size_t off = (size_t)(d - 1) * (N_DIM - 1) - (size_t)(d - 1) * d / 2;
  const float mean = sums[d - 1] / (float)(L * B_DIM);
  float* ob = out + (size_t)b * TOT_PER_B + off;
  for (int k = threadIdx.x; k < L; k += 256) ob[k] -= mean;
}

// ---------------------------------------------------------------------------
extern "C" void kernel_launch(void* const* d_in, const int* in_sizes, int n_in,
                              void* d_out, int out_size, void* d_ws, size_t ws_size,
                              hipStream_t stream)
{
  const float* x  = (const float*)d_in[0];
  const float* wl = (const float*)d_in[1];
  const float* bl = (const float*)d_in[2];
  const float* wr = (const float*)d_in[3];
  const float* br = (const float*)d_in[4];
  const float* wA = (const float*)d_in[5];
  const float* bA = (const float*)d_in[6];
  const float* wB = (const float*)d_in[7];
  const float* bB = (const float*)d_in[8];
  const float* wT = (const float*)d_in[9];
  const float* bT = (const float*)d_in[10];
  const float* wm = (const float*)d_in[11];
  const float* bm = (const float*)d_in[12];
  float* out = (float*)d_out;

  float* ws = (float*)d_ws;
  const size_t rd_elems = (size_t)B_DIM * N_DIM * 16;
  float* RD     = ws;
  float* CD     = RD + rd_elems;
  float* mleft  = CD + rd_elems;
  float* mright = mleft + (size_t)B_DIM * N_DIM;
  float* sums   = mright + (size_t)B_DIM * N_DIM;

  (void)hipMemsetAsync(RD, 0, 2 * rd_elems * sizeof(float), stream);
  (void)hipMemsetAsync(sums, 0, NDIAG * sizeof(float), stream);

  dim3 g1(N_DIM / TS, KSPLIT, B_DIM);
  k_gemm_lr<<<g1, 128, 0, stream>>>(x, wl, wr, RD, CD);

  dim3 g2(N_DIM / CHUNK, B_DIM);
  k_mlp<<<g2, 256, 0, stream>>>(RD, CD, bl, br, wA, bA, wB, bB, wT, bT, wm, bm,
                                mleft, mright);

  dim3 g3(NDIAG, B_DIM);
  k_diag<<<g3, 256, 0, stream>>>(x, mleft, mright, out, sums);
  k_center<<<g3, 256, 0, stream>>>(out, sums);
}